// NodeEncoder_85014582657622
// MI455X (gfx1250) — compile-verified
//
#include <hip/hip_runtime.h>
#include <hip/hip_bf16.h>

#define BB 8
#define SS 1024
#define NN 32
#define HH 128
#define ROWS (BB*SS)          // 8192
#define KF1  133              // H + NLS + 1
#define KF1P 136              // padded to multiple of 4

typedef float v2f __attribute__((ext_vector_type(2)));
typedef float v8f __attribute__((ext_vector_type(8)));
typedef __attribute__((address_space(3))) float lds_float;

__device__ __forceinline__ float leaky(float x) {
    return x > 0.0f ? x : 0.2f * x;
}

__device__ __forceinline__ float wave_sum(float t) {
#pragma unroll
    for (int off = 16; off >= 1; off >>= 1)
        t += __shfl_xor(t, off, 32);
    return t;
}

// ---------------------------------------------------------------------------
// Kernel 1: GAT attention + feature assembly.
// One wave32 per (b,s) row; lane owns channels [4*lane, 4*lane+4).
// Online softmax over 33 entries, single gather pass, 1-deep prefetch.
// ---------------------------------------------------------------------------
__global__ void gat_feats_kernel(const int* __restrict__ subgraph,
                                 const int* __restrict__ neighs,
                                 const float* __restrict__ mask,
                                 const float* __restrict__ local_stats,
                                 const float* __restrict__ global_stats,
                                 const float* __restrict__ emb,
                                 const float* __restrict__ a_w,
                                 const float* __restrict__ a_b,
                                 float* __restrict__ feats)
{
    const int lane = threadIdx.x & 31;
    const int wave = threadIdx.x >> 5;
    const int row  = blockIdx.x * 8 + wave;      // 0..8191
    const int b    = row >> 10;

    const float4 alo = *(const float4*)(a_w + lane * 4);
    const float4 ahi = *(const float4*)(a_w + HH + lane * 4);
    const float  ab  = a_b[0];

    const int src_idx = subgraph[row];
    const float4 src4 = *(const float4*)(emb + (size_t)src_idx * HH + lane * 4);

    float slo = src4.x*alo.x + src4.y*alo.y + src4.z*alo.z + src4.w*alo.w;
    float shi = src4.x*ahi.x + src4.y*ahi.y + src4.z*ahi.z + src4.w*ahi.w;
    slo = wave_sum(slo);
    shi = wave_sum(shi);

    const int   nidx_l = neighs[(size_t)row * NN + lane];
    const float mval_l = mask  [(size_t)row * NN + lane];

    const float s0 = leaky(slo + shi + ab);
    float m = s0;
    float l = 1.0f;
    float4 acc = src4;

    // 1-deep software pipeline over neighbors
    int    idx = __shfl(nidx_l, 0, 32);
    float  mj  = __shfl(mval_l, 0, 32);
    float4 nv  = *(const float4*)(emb + (size_t)idx * HH + lane * 4);

#pragma unroll 4
    for (int j = 0; j < NN; ++j) {
        const float4 cur = nv;
        const float  mc  = mj;
        if (j + 1 < NN) {
            idx = __shfl(nidx_l, j + 1, 32);
            mj  = __shfl(mval_l, j + 1, 32);
            nv  = *(const float4*)(emb + (size_t)idx * HH + lane * 4);
        }
        float t = cur.x*ahi.x + cur.y*ahi.y + cur.z*ahi.z + cur.w*ahi.w;
        t = wave_sum(t);
        const float sc = leaky(slo + t + ab) - 1e9f * mc;
        const float mn = fmaxf(m, sc);
        const float scale = __expf(m - mn);
        const float w  = __expf(sc - mn);
        l = l * scale + w;
        acc.x = fmaf(w, cur.x, acc.x * scale);
        acc.y = fmaf(w, cur.y, acc.y * scale);
        acc.z = fmaf(w, cur.z, acc.z * scale);
        acc.w = fmaf(w, cur.w, acc.w * scale);
        m = mn;
    }

    const float inv = 1.0f / l;
    float4 ne;
    ne.x = acc.x * inv; ne.y = acc.y * inv; ne.z = acc.z * inv; ne.w = acc.w * inv;

    float* fr = feats + (size_t)row * KF1P;
    *(float4*)(fr + lane * 4) = ne;

    if (lane < 8) {
        float v = 0.0f;
        if (lane < 4)       v = local_stats[(size_t)row * 4 + lane];
        else if (lane == 4) v = global_stats[b];
        fr[HH + lane] = v;   // cols 128..135 (133..135 zero-padded)
    }
}

// ---------------------------------------------------------------------------
// Kernels 2/3: out[m][n] = leaky( sum_k A[m][k] * W[n][k] + bias[n] )
// A: [8192 x KP] row-major, W: [128 x WCOLS] row-major.
// Block = 256 threads (8 waves), 128 rows/block; wave w owns M-tile w.
// W staged in LDS 64 output-rows at a time via async global->LDS
// (GLOBAL_LOAD_ASYNC_TO_LDS_B32, ASYNCcnt), padded stride LDW = KP+2,
// zero-filled pad for k in [WCOLS, LDW). Core: V_WMMA_F32_16X16X4_F32
// with all B fragments preloaded to registers (no LDS waits in WMMA chain).
// ---------------------------------------------------------------------------
template<int KP, int LDW, int WCOLS>
__global__ void gemm_leaky_kernel(const float* __restrict__ A,
                                  const float* __restrict__ W,
                                  const float* __restrict__ bias,
                                  float* __restrict__ out)
{
    __shared__ float wsm[64 * LDW];

    const unsigned lbase = (unsigned)(unsigned long long)(lds_float*)wsm;

    const int lane = threadIdx.x & 31;
    const int wave = threadIdx.x >> 5;
    const int half = lane >> 4;      // 0: K=k,k+1   1: K=k+2,k+3
    const int lm   = lane & 15;      // A: row-in-tile; B/out: col-in-tile
    const int m0   = blockIdx.x * 128 + wave * 16;

    for (int nhalf = 0; nhalf < 2; ++nhalf) {
        const int rbase = nhalf * 64;     // first output column in this stage
        __syncthreads();

        // zero the pad columns [WCOLS, LDW) of each LDS row
        for (int i = threadIdx.x; i < 64 * (LDW - WCOLS); i += 256) {
            const int r = i / (LDW - WCOLS);
            const int c = WCOLS + (i - r * (LDW - WCOLS));
            wsm[r * LDW + c] = 0.0f;
        }

        // async copy the valid region straight from global to LDS
        for (int i = threadIdx.x; i < 64 * WCOLS; i += 256) {
            const int r = i / WCOLS;
            const int c = i - r * WCOLS;
            const unsigned ldsOff = lbase + (unsigned)((r * LDW + c) * 4);
            const float* g = W + (size_t)(rbase + r) * WCOLS + c;
            asm volatile("global_load_async_to_lds_b32 %0, %1, off"
                         :: "v"(ldsOff), "v"(g) : "memory");
        }
        asm volatile("s_wait_asynccnt 0x0" ::: "memory");
        __syncthreads();

        for (int nt = 0; nt < 4; ++nt) {
            const int nloc = nt * 16 + lm;          // LDS row (output col - rbase)
            const float* arow = A + (size_t)(m0 + lm) * KP + 2 * half;
            const float* brow = wsm + nloc * LDW + 2 * half;

            // preload all B fragments for this n-tile into registers
            v2f bfr[KP / 4];
#pragma unroll
            for (int kk = 0; kk < KP / 4; ++kk)
                bfr[kk] = *(const v2f*)(brow + kk * 4);

            v8f c = {0.f, 0.f, 0.f, 0.f, 0.f, 0.f, 0.f, 0.f};
#pragma unroll
            for (int kk = 0; kk < KP / 4; ++kk) {
                v2f a = *(const v2f*)(arow + kk * 4);
                c = __builtin_amdgcn_wmma_f32_16x16x4_f32(
                        false, a, false, bfr[kk], (short)0, c, false, false);
            }

            const int ncol = rbase + nt * 16 + lm;
            const float bn = bias[ncol];
#pragma unroll
            for (int r = 0; r < 8; ++r) {           // VGPR r -> rows r / r+8
                const int mrow = m0 + r + 8 * half;
                float v = c[r] + bn;
                v = v > 0.0f ? v : 0.2f * v;
                out[(size_t)mrow * HH + ncol] = v;
            }
        }
    }
}

extern "C" void kernel_launch(void* const* d_in, const int* in_sizes, int n_in,
                              void* d_out, int out_size, void* d_ws, size_t ws_size,
                              hipStream_t stream) {
    const int*   subgraph     = (const int*)  d_in[0];
    const int*   neighs       = (const int*)  d_in[1];
    const float* mask         = (const float*)d_in[2];
    const float* local_stats  = (const float*)d_in[3];
    const float* global_stats = (const float*)d_in[4];
    const float* emb          = (const float*)d_in[5];
    const float* a_w          = (const float*)d_in[6];
    const float* a_b          = (const float*)d_in[7];
    const float* fc1_w        = (const float*)d_in[8];
    const float* fc1_b        = (const float*)d_in[9];
    const float* fc2_w        = (const float*)d_in[10];
    const float* fc2_b        = (const float*)d_in[11];
    float* out = (float*)d_out;

    float* feats = (float*)d_ws;                    // [8192 x 136]
    float* h1    = feats + (size_t)ROWS * KF1P;     // [8192 x 128]

    gat_feats_kernel<<<ROWS / 8, 256, 0, stream>>>(
        subgraph, neighs, mask, local_stats, global_stats, emb, a_w, a_b, feats);

    gemm_leaky_kernel<KF1P, KF1P + 2, KF1><<<ROWS / 128, 256, 0, stream>>>(
        feats, fc1_w, fc1_b, h1);

    gemm_leaky_kernel<HH, HH + 2, HH><<<ROWS / 128, 256, 0, stream>>>(
        h1, fc2_w, fc2_b, out);
}